// ImplicitFunc_Neural_Simplified_65077344469285
// MI455X (gfx1250) — compile-verified
//
#include <hip/hip_runtime.h>
#include <math.h>

typedef float v2f __attribute__((ext_vector_type(2)));
typedef float v8f __attribute__((ext_vector_type(8)));

#define DDIM 128
#define LDP  132   // LDS row pitch (pad to dodge bank conflicts)

// Compute one 16x16 f32 tile of  OUT = A(16x128, in LDS) x B(128x16, from W)
// using V_WMMA_F32_16X16X4_F32, looping K in steps of 4.
// wT==true : B[k][n] = W[n*128+k]  (i.e. OUT = A @ W^T)
// wT==false: B[k][n] = W[k*128+n]  (i.e. OUT = A @ W)
// f32 WMMA VGPR layout (ISA 7.12.2): A vgpr0 = {K=0 | K=2}, vgpr1 = {K=1 | K=3}
// across lane halves; C vgpr r = row r (lanes 0-15) / row r+8 (lanes 16-31).
__device__ __forceinline__ v8f gemm_tile_f32(const float* __restrict__ Als,
                                             const float* __restrict__ W,
                                             int n0, int lane, bool wT)
{
    const int half = lane >> 4;   // which half-wave
    const int m    = lane & 15;   // row (for A) / col (for B) within tile
    v8f c = {};
    #pragma unroll 4
    for (int k = 0; k < DDIM; k += 4) {
        const int ka = k + 2 * half;
        v2f a, b;
        a.x = Als[m * LDP + ka];
        a.y = Als[m * LDP + ka + 1];
        const int n = n0 + m;
        if (wT) { b.x = W[n * DDIM + ka];     b.y = W[n * DDIM + ka + 1]; }
        else    { b.x = W[ka * DDIM + n];     b.y = W[(ka + 1) * DDIM + n]; }
        c = __builtin_amdgcn_wmma_f32_16x16x4_f32(false, a, false, b,
                                                  (short)0, c, false, false);
    }
    return c;
}

// out <- x  (so edge kernel can atomically accumulate -Phi*diff straight in),
// degree <- 0
__global__ void init_kernel(const float* __restrict__ x, float* __restrict__ out,
                            float* __restrict__ degree, int nNodes)
{
    int i = blockIdx.x * blockDim.x + threadIdx.x;
    if (i < nNodes * DDIM) out[i] = x[i];
    if (i < nNodes)        degree[i] = 0.0f;
}

__global__ void degree_kernel(const long long* __restrict__ ei,
                              const float* __restrict__ ew,
                              float* __restrict__ degree, int nEdges)
{
    int e = blockIdx.x * blockDim.x + threadIdx.x;
    // hardware GLOBAL_ATOMIC_ADD_F32, no-return (STOREcnt), no CAS loop
    if (e < nEdges) unsafeAtomicAdd(&degree[(int)ei[e]], ew[e]);
}

// Per 16-row tile: z = x@Wchi^T ; phi_chi = tanh(||z||) ; y = z@Wphi^T ;
// v = x@Wvar^T ; u = v@Wvar ; q = rowdot(x,u).
__global__ void node_kernel(const float* __restrict__ x,
                            const float* __restrict__ Wchi,
                            const float* __restrict__ Wphi,
                            const float* __restrict__ Wvar,
                            float* __restrict__ y, float* __restrict__ u,
                            float* __restrict__ phi_chi, float* __restrict__ q,
                            int nNodes)
{
    __shared__ float xs[16 * LDP];
    __shared__ float zs[16 * LDP];
    __shared__ float us[16 * LDP];

    const int tid  = threadIdx.x;
    const int lane = tid & 31;
    const int wav  = tid >> 5;          // 8 waves -> 8 column tiles
    const int row0 = blockIdx.x * 16;
    const int n0   = wav * 16;
    const int half = lane >> 4;
    const int m    = lane & 15;

    // stage x tile (clamped reads keep every lane active for WMMA)
    for (int i = tid; i < 16 * DDIM; i += 256) {
        int r  = i >> 7, cc = i & 127;
        int rr = row0 + r; if (rr >= nNodes) rr = nNodes - 1;
        xs[r * LDP + cc] = x[(size_t)rr * DDIM + cc];
    }
    __syncthreads();

    // z = x @ Wchi^T  -> zs
    {
        v8f c = gemm_tile_f32(xs, Wchi, n0, lane, true);
        #pragma unroll
        for (int r = 0; r < 8; ++r) zs[(r + 8 * half) * LDP + n0 + m] = c[r];
    }
    __syncthreads();

    // phi_chi = tanh(||z_row||)
    if (tid < 16 && row0 + tid < nNodes) {
        float s = 0.0f;
        #pragma unroll 8
        for (int k = 0; k < DDIM; ++k) { float zv = zs[tid * LDP + k]; s += zv * zv; }
        phi_chi[row0 + tid] = tanhf(sqrtf(s));
    }

    // y = z @ Wphi^T  -> global
    {
        v8f c = gemm_tile_f32(zs, Wphi, n0, lane, true);
        #pragma unroll
        for (int r = 0; r < 8; ++r) {
            int rr = row0 + r + 8 * half;
            if (rr < nNodes) y[(size_t)rr * DDIM + n0 + m] = c[r];
        }
    }
    __syncthreads();                       // done reading zs

    // v = x @ Wvar^T  -> zs (reuse)
    {
        v8f c = gemm_tile_f32(xs, Wvar, n0, lane, true);
        #pragma unroll
        for (int r = 0; r < 8; ++r) zs[(r + 8 * half) * LDP + n0 + m] = c[r];
    }
    __syncthreads();

    // u = v @ Wvar  -> us + global
    {
        v8f c = gemm_tile_f32(zs, Wvar, n0, lane, false);
        #pragma unroll
        for (int r = 0; r < 8; ++r) {
            int rr = row0 + r + 8 * half;
            us[(r + 8 * half) * LDP + n0 + m] = c[r];
            if (rr < nNodes) u[(size_t)rr * DDIM + n0 + m] = c[r];
        }
    }
    __syncthreads();

    // q = rowdot(x, u)
    if (tid < 16 && row0 + tid < nNodes) {
        float s = 0.0f;
        #pragma unroll 8
        for (int k = 0; k < DDIM; ++k) s += xs[tid * LDP + k] * us[tid * LDP + k];
        q[row0 + tid] = s;
    }
}

// One wave32 per edge: two 128-length dots (float4 per lane + shfl_xor tree),
// then 4 hardware f32 atomics per lane into out (accumulating -Phi*(x_r-x_c)).
__global__ void edge_kernel(const float* __restrict__ x,
                            const float* __restrict__ y,
                            const float* __restrict__ u,
                            const float* __restrict__ phi_chi,
                            const float* __restrict__ q,
                            const float* __restrict__ degree,
                            const long long* __restrict__ ei,
                            const float* __restrict__ ew,
                            float* __restrict__ out, int nEdges)
{
    const int gtid = blockIdx.x * blockDim.x + threadIdx.x;
    const int e    = gtid >> 5;
    const int lane = gtid & 31;
    if (e >= nEdges) return;

    const int r = (int)ei[e];
    const int c = (int)ei[nEdges + e];
    const int base = lane * 4;

    const float4 xr = *(const float4*)&x[(size_t)r * DDIM + base];
    const float4 xc = *(const float4*)&x[(size_t)c * DDIM + base];
    const float4 yr = *(const float4*)&y[(size_t)r * DDIM + base];
    const float4 yc = *(const float4*)&y[(size_t)c * DDIM + base];
    const float4 uc = *(const float4*)&u[(size_t)c * DDIM + base];

    float p1 = yr.x * yc.x + yr.y * yc.y + yr.z * yc.z + yr.w * yc.w; // y_r . y_c
    float p2 = xr.x * uc.x + xr.y * uc.y + xr.z * uc.z + xr.w * uc.w; // x_r . u_c
    #pragma unroll
    for (int off = 16; off >= 1; off >>= 1) {
        p1 += __shfl_xor(p1, off, 32);
        p2 += __shfl_xor(p2, off, 32);
    }

    const float dmd  = q[r] + q[c] - 2.0f * p2;            // ||diff @ Wv^T||^2
    const float nrm  = sqrtf(fmaxf(dmd, 0.0f));
    const float phiv = tanhf(1.0f / (nrm + 1e-6f));
    const float phip = tanhf(fabsf(p1));
    const float Phi  = ew[e] / degree[r] * phip * phiv / phi_chi[r];
    const float s    = -Phi;                                // MU = 1

    float* o = &out[(size_t)r * DDIM + base];
    // hardware GLOBAL_ATOMIC_ADD_F32 (no return value -> no LOADcnt stall)
    unsafeAtomicAdd(o + 0, s * (xr.x - xc.x));
    unsafeAtomicAdd(o + 1, s * (xr.y - xc.y));
    unsafeAtomicAdd(o + 2, s * (xr.z - xc.z));
    unsafeAtomicAdd(o + 3, s * (xr.w - xc.w));
}

extern "C" void kernel_launch(void* const* d_in, const int* in_sizes, int n_in,
                              void* d_out, int out_size, void* d_ws, size_t ws_size,
                              hipStream_t stream)
{
    const float*     x    = (const float*)d_in[0];
    const long long* ei   = (const long long*)d_in[1];   // int64 edge_index [2,E]
    const float*     ew   = (const float*)d_in[2];
    const float*     Wchi = (const float*)d_in[3];
    const float*     Wphi = (const float*)d_in[4];
    const float*     Wvar = (const float*)d_in[5];
    float*           out  = (float*)d_out;

    const int N = in_sizes[0] / DDIM;
    const int E = in_sizes[2];

    // workspace layout (floats): degree[N] | phi_chi[N] | q[N] | y[N*D] | u[N*D]
    float* ws      = (float*)d_ws;
    float* degree  = ws;
    float* phi_chi = ws + N;
    float* q       = ws + 2 * (size_t)N;
    float* y       = ws + 3 * (size_t)N;
    float* u       = y + (size_t)N * DDIM;

    init_kernel  <<<(N * DDIM + 255) / 256, 256, 0, stream>>>(x, out, degree, N);
    degree_kernel<<<(E + 255) / 256,        256, 0, stream>>>(ei, ew, degree, E);
    node_kernel  <<<(N + 15) / 16,          256, 0, stream>>>(x, Wchi, Wphi, Wvar,
                                                              y, u, phi_chi, q, N);
    // one wave32 per edge
    const long long ethreads = (long long)E * 32;
    edge_kernel  <<<(int)((ethreads + 255) / 256), 256, 0, stream>>>(
        x, y, u, phi_chi, q, degree, ei, ew, out, E);
}